// EnhancedSpatialAttention_13451837571349
// MI455X (gfx1250) — compile-verified
//
#include <hip/hip_runtime.h>
#include <math.h>

typedef __attribute__((ext_vector_type(2))) float v2f;
typedef __attribute__((ext_vector_type(8))) float v8f;

#define NT 512
#define KEPS 1e-8f
#define KSLOPE 0.2f

// ---------------------------------------------------------------------------
// Kernel 1: h = X @ W_proj  ([V,F] x [F,16] -> [V,16]) via V_WMMA_F32_16X16X4_F32
// One wave (32 lanes) per 16-row tile. N dim == H == 16 exactly.
// A 16x4 layout: lane m=lane&15, VGPR v holds K = v + 2*(lane>>4)
// B 4x16 layout: VGPR v holds row K = v + 2*(lane>>4), col N = lane&15
// C/D 16x16:     VGPR v holds row M = v + 8*(lane>>4), col N = lane&15
// ---------------------------------------------------------------------------
__global__ __launch_bounds__(32) void proj_wmma_kernel(
    const float* __restrict__ X, const float* __restrict__ Wp,
    float* __restrict__ h, int V, int F)
{
  const int lane = threadIdx.x;
  const int m    = lane & 15;
  const int half = lane >> 4;
  const int n    = lane & 15;
  const int i0   = blockIdx.x * 16;
  v8f acc = {};
  for (int k = 0; k < F; k += 4) {
    const int kc = k + 2 * half;
    v2f a, b;
    const float* arow = X + (size_t)(i0 + m) * F + kc;
    a.x = arow[0];
    a.y = arow[1];
    b.x = Wp[(size_t)kc * 16 + n];
    b.y = Wp[(size_t)(kc + 1) * 16 + n];
    acc = __builtin_amdgcn_wmma_f32_16x16x4_f32(false, a, false, b,
                                                (short)0, acc, false, false);
  }
#pragma unroll
  for (int v = 0; v < 8; ++v)
    h[(size_t)(i0 + v + 8 * half) * 16 + n] = acc[v];
}

// ---------------------------------------------------------------------------
// Kernel 2: per-node aux: [px,py,vx,vy,dirx,diry,li,lj] (8 floats) + group idx
// ---------------------------------------------------------------------------
__global__ void aux_kernel(
    const float* __restrict__ pos, const float* __restrict__ vel,
    const int* __restrict__ types, const int* __restrict__ pnum,
    const float* __restrict__ h, const float* __restrict__ wsc,
    float* __restrict__ aux, int* __restrict__ gc, int V, int H, int G)
{
  int i = blockIdx.x * blockDim.x + threadIdx.x;
  if (i >= V) return;
  float px = pos[2 * i], py = pos[2 * i + 1];
  float vx = vel[2 * i], vy = vel[2 * i + 1];
  float speed = sqrtf(vx * vx + vy * vy);
  float inv = 1.0f / (speed + KEPS);
  float li = 0.f, lj = 0.f;
  for (int c = 0; c < H; ++c) {
    float hv = h[(size_t)i * H + c];
    li += hv * wsc[c];
    lj += hv * wsc[H + c];
  }
  float* a = aux + (size_t)i * 8;
  a[0] = px; a[1] = py; a[2] = vx; a[3] = vy;
  a[4] = vx * inv; a[5] = vy * inv; a[6] = li; a[7] = lj;
  int g = i - pnum[0];
  gc[i] = (types[i] == 1 && g >= 0 && g < G) ? g : -1;
}

// ---------------------------------------------------------------------------
// Block reductions over NT threads
// ---------------------------------------------------------------------------
__device__ __forceinline__ float blockMaxF(float v, float* red)
{
  int tid = threadIdx.x;
  red[tid] = v;
  __syncthreads();
  for (int s = NT / 2; s > 0; s >>= 1) {
    if (tid < s) red[tid] = fmaxf(red[tid], red[tid + s]);
    __syncthreads();
  }
  float r = red[0];
  __syncthreads();
  return r;
}

__device__ __forceinline__ float blockSumF(float v, float* red)
{
  int tid = threadIdx.x;
  red[tid] = v;
  __syncthreads();
  for (int s = NT / 2; s > 0; s >>= 1) {
    if (tid < s) red[tid] += red[tid + s];
    __syncthreads();
  }
  float r = red[0];
  __syncthreads();
  return r;
}

// ---------------------------------------------------------------------------
// Kernel 3: fused per-row phi + logits + masked softmax + renormalized attn.
// One 512-thread block per row; CHUNKS j-values per thread held in registers
// across the three block reductions. phi stored as coalesced float4.
// ---------------------------------------------------------------------------
template <int CHUNKS>
__global__ __launch_bounds__(NT) void row_kernel(
    const float* __restrict__ aux, const int* __restrict__ gc,
    const float* __restrict__ adjm, const float* __restrict__ conf,
    const float* __restrict__ wsc, float* __restrict__ phi,
    float* __restrict__ attn, int V, int H, int G)
{
  __shared__ float red[NT];
  const int i = blockIdx.x;
  const int tid = threadIdx.x;
  // row-uniform scalars (blockIdx-only dependence -> scalar loads)
  const float* ai = aux + (size_t)i * 8;
  const float pix = ai[0], piy = ai[1], vix = ai[2], viy = ai[3];
  const float dix = ai[4], diy = ai[5], li = ai[6];
  const int gci = gc[i];
  const int wb = 2 * H;
  const float w0 = wsc[wb], w1 = wsc[wb + 1], w2 = wsc[wb + 2], w3 = wsc[wb + 3];
  const float* adjrow = adjm + (size_t)i * V;

  float ml[CHUNKS];  // masked logits, later reused to hold exp values
  float av[CHUNKS];  // adjacency values, later reused to hold attn*adj/S
#pragma unroll
  for (int c = 0; c < CHUNKS; ++c) {
    int j = c * NT + tid;
    ml[c] = -INFINITY;
    av[c] = 0.f;
    if (j < V) {
      const float4* a4 = (const float4*)(aux + (size_t)j * 8);
      float4 p0 = a4[0];  // px, py, vx, vy
      float4 p1 = a4[1];  // dirx, diry, li_j, lj_j
      float dx = pix - p0.x, dy = piy - p0.y;
      float dist = sqrtf(dx * dx + dy * dy);
      float dvx = vix - p0.z, dvy = viy - p0.w;
      float vd = sqrtf(dvx * dvx + dvy * dvy);
      float da = fminf(1.f, fmaxf(-1.f, dix * p1.x + diy * p1.y));
      int gcj = gc[j];
      float nc = (gci >= 0 && gcj >= 0) ? conf[(size_t)gci * G + gcj] : 0.f;
      float4 ph;
      ph.x = dist; ph.y = vd; ph.z = da; ph.w = nc;
      *(float4*)(phi + ((size_t)i * V + j) * 4) = ph;   // coalesced 16B store
      float adj = adjrow[j];
      float logit = li + p1.w + dist * w0 + vd * w1 + da * w2 + nc * w3;
      logit = (logit >= 0.f) ? logit : KSLOPE * logit;  // LeakyReLU first
      ml[c] = (adj > 0.f) ? logit : logit - 1e9f;       // then suppress mask
      av[c] = adj;
    }
  }

  // softmax over full row of masked logits
  float lm = -INFINITY;
#pragma unroll
  for (int c = 0; c < CHUNKS; ++c) lm = fmaxf(lm, ml[c]);
  const float M = blockMaxF(lm, red);

  float ls = 0.f;
#pragma unroll
  for (int c = 0; c < CHUNKS; ++c) {
    int j = c * NT + tid;
    float e = (j < V) ? __expf(ml[c] - M) : 0.f;
    ml[c] = e;
    ls += e;
  }
  const float S = blockSumF(ls, red);
  const float invS = 1.f / S;

  // a = softmax * adjacency; renormalize by (sum(a) + eps)
  float lt = 0.f;
#pragma unroll
  for (int c = 0; c < CHUNKS; ++c) {
    av[c] = ml[c] * av[c] * invS;
    lt += av[c];
  }
  const float invd = 1.f / (blockSumF(lt, red) + KEPS);
#pragma unroll
  for (int c = 0; c < CHUNKS; ++c) {
    int j = c * NT + tid;
    if (j < V) attn[(size_t)i * V + j] = av[c] * invd;
  }
}

// ---------------------------------------------------------------------------
// Kernel 4: attended = attn @ h  ([V,V] x [V,16]) via V_WMMA_F32_16X16X4_F32.
// One wave per 16-row tile; attn A-tile (16x64) staged through LDS with
// stride-68 padding (bank-conflict-free), 256B contiguous global loads.
// h is 256KB -> L2-resident, B operand read directly from global.
// ---------------------------------------------------------------------------
__global__ __launch_bounds__(32) void attend_wmma_kernel(
    const float* __restrict__ attn, const float* __restrict__ h,
    float* __restrict__ out, int V)
{
  __shared__ float At[16][68];
  const int lane = threadIdx.x;
  const int m    = lane & 15;
  const int half = lane >> 4;
  const int n    = lane & 15;
  const int i0   = blockIdx.x * 16;
  v8f acc = {};
  for (int kb = 0; kb < V; kb += 64) {
    // cooperative load of 16x64 attn tile: each half-wave covers one row chunk
#pragma unroll
    for (int rep = 0; rep < 8; ++rep) {
      int idx = rep * 128 + lane * 4;
      int r = idx >> 6, c = idx & 63;
      float4 v = *(const float4*)(attn + (size_t)(i0 + r) * V + kb + c);
      *(float4*)&At[r][c] = v;
    }
    __syncthreads();
#pragma unroll
    for (int kk = 0; kk < 16; ++kk) {
      const int kcol = kk * 4 + 2 * half;
      v2f a, b;
      a.x = At[m][kcol];
      a.y = At[m][kcol + 1];
      const size_t krow = (size_t)(kb + kcol);
      b.x = h[krow * 16 + n];
      b.y = h[(krow + 1) * 16 + n];
      acc = __builtin_amdgcn_wmma_f32_16x16x4_f32(false, a, false, b,
                                                  (short)0, acc, false, false);
    }
    __syncthreads();
  }
#pragma unroll
  for (int v = 0; v < 8; ++v)
    out[(size_t)(i0 + v + 8 * half) * 16 + n] = acc[v];
}

// ---------------------------------------------------------------------------
extern "C" void kernel_launch(void* const* d_in, const int* in_sizes, int n_in,
                              void* d_out, int out_size, void* d_ws, size_t ws_size,
                              hipStream_t stream)
{
  const float* X    = (const float*)d_in[0];  // node_features [V,F]
  const float* adj  = (const float*)d_in[1];  // adjacency [V,V]
  const float* pos  = (const float*)d_in[2];  // positions [V,2]
  const float* vel  = (const float*)d_in[3];  // velocities [V,2]
  const int*   typ  = (const int*)d_in[4];    // node_types [V]
  const int*   pnum = (const int*)d_in[5];    // num_ped_nodes scalar
  const float* conf = (const float*)d_in[6];  // conflict matrix [G,G]
  const float* Wp   = (const float*)d_in[7];  // W_proj [F,H]
  const float* wsc  = (const float*)d_in[8];  // w_score [2H+4]

  const int V = in_sizes[2] / 2;
  const int F = in_sizes[0] / V;
  const int H = in_sizes[7] / F;          // == 16
  int G = 1;
  while ((long long)(G + 1) * (G + 1) <= (long long)in_sizes[6]) ++G;

  // output layout: attended [V,H] | attn [V,V] | phi [V,V,4]
  float* outp     = (float*)d_out;
  float* attended = outp;
  float* attnp    = outp + (size_t)V * H;
  float* phip     = attnp + (size_t)V * V;

  // workspace: h [V,16] | aux [V,8] | gc [V]
  float* hbuf = (float*)d_ws;
  float* aux  = hbuf + (size_t)V * 16;
  int*   gc   = (int*)(aux + (size_t)V * 8);

  proj_wmma_kernel<<<V / 16, 32, 0, stream>>>(X, Wp, hbuf, V, F);
  aux_kernel<<<(V + 255) / 256, 256, 0, stream>>>(pos, vel, typ, pnum, hbuf,
                                                  wsc, aux, gc, V, H, G);

  const int chunks = (V + NT - 1) / NT;
  if (chunks <= 1)
    row_kernel<1><<<V, NT, 0, stream>>>(aux, gc, adj, conf, wsc, phip, attnp, V, H, G);
  else if (chunks <= 2)
    row_kernel<2><<<V, NT, 0, stream>>>(aux, gc, adj, conf, wsc, phip, attnp, V, H, G);
  else if (chunks <= 4)
    row_kernel<4><<<V, NT, 0, stream>>>(aux, gc, adj, conf, wsc, phip, attnp, V, H, G);
  else if (chunks <= 8)
    row_kernel<8><<<V, NT, 0, stream>>>(aux, gc, adj, conf, wsc, phip, attnp, V, H, G);
  else if (chunks <= 16)
    row_kernel<16><<<V, NT, 0, stream>>>(aux, gc, adj, conf, wsc, phip, attnp, V, H, G);
  else
    row_kernel<32><<<V, NT, 0, stream>>>(aux, gc, adj, conf, wsc, phip, attnp, V, H, G);

  attend_wmma_kernel<<<V / 16, 32, 0, stream>>>(attnp, hbuf, attended, V);
}